// NeighbourPCA_38285338476775
// MI455X (gfx1250) — compile-verified
//
#include <hip/hip_runtime.h>

typedef __attribute__((ext_vector_type(2))) float v2f;
typedef __attribute__((ext_vector_type(8))) float v8f;

#define VN 50000
#define KN 32
#define FN 16
#define WPB 4
#define NSWEEPS 7

__global__ __launch_bounds__(WPB * 32)
void NeighbourPCA_kernel(const float* __restrict__ feat,
                         const int*   __restrict__ nidx,
                         float*       __restrict__ out) {
  __shared__ __align__(16) float ldsG[WPB][FN * FN];

  const int tid  = threadIdx.x;
  const int wave = tid >> 5;
  const int lane = tid & 31;
  const int node = blockIdx.x * WPB + wave;
  if (node >= VN) return;                 // wave-uniform: EXEC all-1s at WMMA

  const int l16 = lane & 15;
  const int hi  = lane >> 4;              // 0 = G rows, 1 = V rows

  // ---------------- G = A^T * A via 8x v_wmma_f32_16x16x4_f32 ----------------
  // A-operand layout of A^T (16x4) == B-operand layout of A (4x16) elementwise,
  // so one gathered register pair feeds both SRC0 and SRC1.
  v8f acc = {};
  const int* nb = nidx + node * KN;
  #pragma unroll
  for (int c = 0; c < 8; ++c) {
    const int k0 = 4 * c + 2 * hi;
    const int n0 = nb[k0];
    const int n1 = nb[k0 + 1];
    v2f a;
    a.x = feat[n0 * FN + l16];
    a.y = feat[n1 * FN + l16];
    acc = __builtin_amdgcn_wmma_f32_16x16x4_f32(
        false, a, false, a, (short)0, acc, false, false);
  }

  // WMMA C/D layout -> row-major G in LDS (lane l, reg r holds G[r+8*hi][l16])
  float* g = ldsG[wave];
  #pragma unroll
  for (int r = 0; r < 8; ++r)
    g[(r + 8 * hi) * FN + l16] = acc[r];

  // ---- all lanes load row l16 unconditionally (no exec-divergent ds loads),
  //      then blend: lower half = G row, upper half = identity row (V init).
  float m[16];
  {
    const float4* grow = (const float4*)(g + l16 * FN);
    const float4 t0 = grow[0], t1 = grow[1], t2 = grow[2], t3 = grow[3];
    const float gv[16] = {t0.x, t0.y, t0.z, t0.w, t1.x, t1.y, t1.z, t1.w,
                          t2.x, t2.y, t2.z, t2.w, t3.x, t3.y, t3.z, t3.w};
    #pragma unroll
    for (int j = 0; j < 16; ++j)
      m[j] = hi ? ((j == l16) ? 1.0f : 0.0f) : gv[j];
  }

  // -------- two-sided parallel Jacobi: G <- J^T G J ; V <- V J ----------
  // lanes 0-15: rows of G (column update + row update)
  // lanes 16-31: rows of V (column update only)
  #pragma unroll 1
  for (int sweep = 0; sweep < NSWEEPS; ++sweep) {
    #pragma unroll
    for (int r = 0; r < 15; ++r) {
      float cs[8], sn[8];

      // rotation parameters for all 8 disjoint pairs (3 broadcasts each)
      #pragma unroll
      for (int k = 0; k < 8; ++k) {
        const int aa = (k == 0) ? 15 : (r + k) % 15;
        const int bb = (k == 0) ? r  : (r - k + 15) % 15;
        const int p  = (aa < bb) ? aa : bb;
        const int q  = (aa < bb) ? bb : aa;

        const float app = __shfl(m[p], p, 32);   // G[p][p] from lane p
        const float aqq = __shfl(m[q], q, 32);   // G[q][q] from lane q
        const float apq = __shfl(m[q], p, 32);   // G[p][q] from lane p

        const float th = (aqq - app) * 0.5f / apq;
        float t  = copysignf(1.0f, th) / (fabsf(th) + sqrtf(1.0f + th * th));
        float c_ = rsqrtf(1.0f + t * t);
        float s_ = t * c_;
        const bool skip = !(fabsf(apq) > 1e-30f);  // also kills 0/0 NaN path
        cs[k] = skip ? 1.0f : c_;
        sn[k] = skip ? 0.0f : s_;
      }

      // column update (G J and V J) — lane-local on all 32 lanes
      #pragma unroll
      for (int k = 0; k < 8; ++k) {
        const int aa = (k == 0) ? 15 : (r + k) % 15;
        const int bb = (k == 0) ? r  : (r - k + 15) % 15;
        const int p  = (aa < bb) ? aa : bb;
        const int q  = (aa < bb) ? bb : aa;
        const float x = m[p], y = m[q];
        m[p] = cs[k] * x - sn[k] * y;
        m[q] = sn[k] * x + cs[k] * y;
      }

      // row update (J^T applied to G): pick this lane's pair coefficients
      float cl = 1.0f, sl = 0.0f;
      #pragma unroll
      for (int k = 0; k < 8; ++k) {
        const int aa = (k == 0) ? 15 : (r + k) % 15;
        const int bb = (k == 0) ? r  : (r - k + 15) % 15;
        const int p  = (aa < bb) ? aa : bb;
        const int q  = (aa < bb) ? bb : aa;
        const bool isp = (l16 == p);
        const bool isq = (l16 == q);
        cl = (isp || isq) ? cs[k] : cl;
        sl = isp ? -sn[k] : (isq ? sn[k] : sl);
      }
      cl = hi ? 1.0f : cl;                 // V rows: no row update
      sl = hi ? 0.0f : sl;

      // partner lane for this round (closed form of the tournament schedule)
      int pr = (l16 == 15) ? r : ((l16 == r) ? 15 : ((2 * r + 30 - l16) % 15));
      pr += hi << 4;                        // keep upper half self-contained

      // one full-row exchange serves all 8 pairs simultaneously
      #pragma unroll
      for (int j = 0; j < 16; ++j) {
        const float prj = __shfl(m[j], pr, 32);
        m[j] = cl * m[j] + sl * prj;
      }
    }
  }

  // ---------------- eigenvalues (diagonal), sort, scaled output ----------------
  float lam = 0.0f;
  #pragma unroll
  for (int j = 0; j < 16; ++j)
    lam = (l16 == j) ? m[j] : lam;          // lane i (lower half) holds G[i][i]

  float ev[16];
  #pragma unroll
  for (int j = 0; j < 16; ++j)
    ev[j] = __shfl(lam, j, 32);             // broadcast from lanes 0-15 to all

  // stable descending rank of each original column
  int rank[16];
  #pragma unroll
  for (int c = 0; c < 16; ++c) {
    int rk = 0;
    #pragma unroll
    for (int k = 0; k < 16; ++k)
      rk += (ev[k] > ev[c]) || ((ev[k] == ev[c]) && (k < c));
    rank[c] = rk;
  }

  // this lane's row index i needs the i-th largest eigenvalue
  float sval = 0.0f;
  #pragma unroll
  for (int c = 0; c < 16; ++c)
    sval = (rank[c] == l16) ? ev[c] : sval;

  const float sigma = sqrtf(fmaxf(sval, 0.0f));   // singular value of A
  const float scale = sqrtf(sigma + 1e-6f);       // sqrt(s + EPS)

  // upper half holds rows of eigenvector matrix V; out[i*16 + rank[c]] = scale_i * V[i][c]
  if (hi) {
    float* o = out + (size_t)node * (FN * FN) + (size_t)l16 * FN;
    #pragma unroll
    for (int c = 0; c < 16; ++c)
      o[rank[c]] = scale * m[c];
  }
}

extern "C" void kernel_launch(void* const* d_in, const int* in_sizes, int n_in,
                              void* d_out, int out_size, void* d_ws, size_t ws_size,
                              hipStream_t stream) {
  const float* feat = (const float*)d_in[0];
  const int*   nidx = (const int*)d_in[1];
  float*       out  = (float*)d_out;

  const int grid = (VN + WPB - 1) / WPB;
  hipLaunchKernelGGL(NeighbourPCA_kernel, dim3(grid), dim3(WPB * 32),
                     0, stream, feat, nidx, out);
}